// MonotonicityLoss_33612414058531
// MI455X (gfx1250) — compile-verified
//
#include <hip/hip_runtime.h>
#include <hip/hip_bf16.h>

typedef __attribute__((ext_vector_type(2))) float v2f;
typedef __attribute__((ext_vector_type(8))) float v8f;

#define MARGIN 0.1f

// ---------------------------------------------------------------------------
// Kernel 0: zero the workspace (gains[B] + {total, count}) every launch.
// ---------------------------------------------------------------------------
__global__ __launch_bounds__(256) void zero_ws_kernel(float* __restrict__ ws, int n) {
    int i = blockIdx.x * 256 + threadIdx.x;
    if (i < n) ws[i] = 0.0f;
}

// ---------------------------------------------------------------------------
// Kernel 1: row-sum reduction of g (B x K, f32) into gains[B].
// One wave (32 threads) per block. Block (gx, gy) owns rows
// [gx*16, gx*16+16) and K-slice [gy*512, gy*512+512).
//
// Lane L handles row r0 + (L & 15). Lanes 0-15 take the first 256 K of the
// slice, lanes 16-31 the second 256 K, each as 64 contiguous float4 loads
// (global_load_b128 streams -> this is the HBM-draining loop, ~5.8 us floor
// for 134 MB at 23.3 TB/s).
//
// The wave's 32 lanes x 2 folded partials form a 16x4 f32 A-matrix
// (2 VGPRs, matching the CDNA5 16x4 f32 A layout). One
// V_WMMA_F32_16X16X4_F32 against an all-ones B reduces it to 16 row sums:
// D[m][n] = sum_k A[m][k] for every n. Per the C/D layout, VGPR j holds
// row j on lanes 0-15 and row j+8 on lanes 16-31; lanes 0 and 16 commit the
// 16 partial row sums with f32 global atomics.
// ---------------------------------------------------------------------------
__global__ __launch_bounds__(32) void gains_wmma_kernel(const float* __restrict__ g,
                                                        float* __restrict__ gains,
                                                        int K) {
    const int lane = threadIdx.x;        // 0..31, full wave32
    const int l16  = lane & 15;
    const int half = lane >> 4;
    const int r0   = blockIdx.x << 4;    // first row of this 16-row group

    const long long row = (long long)(r0 + l16);
    const float4* p = (const float4*)(g + row * (long long)K
                                        + (long long)blockIdx.y * 512
                                        + half * 256);

    float ax = 0.0f, ay = 0.0f, az = 0.0f, aw = 0.0f;
#pragma unroll 8
    for (int i = 0; i < 64; ++i) {       // 256 floats = 64 x b128 per lane
        float4 v = p[i];
        ax += v.x; ay += v.y; az += v.z; aw += v.w;
    }

    // Fold 4 partials -> the 2 A-matrix VGPRs this lane owns.
    v2f a; a.x = ax + az; a.y = ay + aw;
    // All-ones B: D[m][n] = sum over the 4 K slots of row m.
    v2f b; b.x = 1.0f; b.y = 1.0f;
    v8f c = {};
    // (neg_a, A, neg_b, B, c_mod, C, reuse_a, reuse_b)
    c = __builtin_amdgcn_wmma_f32_16x16x4_f32(
            false, a, false, b, (short)0, c, false, false);

    // VGPR j: lane 0 -> row r0+j, lane 16 -> row r0+8+j (N replicated).
#pragma unroll
    for (int j = 0; j < 8; ++j) {
        if (lane == 0)  atomicAdd(&gains[r0 + j],     c[j]);
        if (lane == 16) atomicAdd(&gains[r0 + 8 + j], c[j]);
    }
}

// ---------------------------------------------------------------------------
// Kernel 2: pairwise hinge. Block i scans j in (i, B); mask is
// (norms[i] > norms[j]) since blockIdx already enforces i < j.
// gains/norms are 16 KB each -> cache resident; compute is trivial.
// ---------------------------------------------------------------------------
__global__ __launch_bounds__(256) void pair_loss_kernel(const float* __restrict__ norms,
                                                        const float* __restrict__ gains,
                                                        float* __restrict__ acc,
                                                        int B) {
    __shared__ float st[256];
    __shared__ float sc[256];

    const int i  = blockIdx.x;
    const float ni = norms[i];
    const float gi = gains[i];

    float t = 0.0f, cnt = 0.0f;
    for (int j = i + 1 + (int)threadIdx.x; j < B; j += 256) {
        if (ni > norms[j]) {
            float d = gains[j] - gi + MARGIN;
            t   += fmaxf(d, 0.0f);
            cnt += 1.0f;
        }
    }
    st[threadIdx.x] = t;
    sc[threadIdx.x] = cnt;
    __syncthreads();
#pragma unroll
    for (int s = 128; s > 0; s >>= 1) {
        if ((int)threadIdx.x < s) {
            st[threadIdx.x] += st[threadIdx.x + s];
            sc[threadIdx.x] += sc[threadIdx.x + s];
        }
        __syncthreads();
    }
    if (threadIdx.x == 0) {
        atomicAdd(&acc[0], st[0]);
        atomicAdd(&acc[1], sc[0]);
    }
}

// ---------------------------------------------------------------------------
// Kernel 3: finalize -> out[0] = count > 0 ? total / max(count,1) : 0
// ---------------------------------------------------------------------------
__global__ void finalize_kernel(const float* __restrict__ acc, float* __restrict__ out) {
    float t = acc[0];
    float c = acc[1];
    out[0] = (c > 0.0f) ? (t / fmaxf(c, 1.0f)) : 0.0f;
}

extern "C" void kernel_launch(void* const* d_in, const int* in_sizes, int n_in,
                              void* d_out, int out_size, void* d_ws, size_t ws_size,
                              hipStream_t stream) {
    const float* norms = (const float*)d_in[0];   // (B,)    f32
    const float* g     = (const float*)d_in[1];   // (B,H,N) f32 flat
    float* out = (float*)d_out;                   // scalar f32

    const int B = in_sizes[0];                    // 4096
    const int K = (int)((long long)in_sizes[1] / (long long)B); // H*N = 8192

    float* gains = (float*)d_ws;                  // [B]
    float* acc   = gains + B;                     // [2] = {total, count}

    const int nz = B + 2;
    zero_ws_kernel<<<(nz + 255) / 256, 256, 0, stream>>>(gains, nz);

    // B/16 row groups x K/512 K-slices, one wave each. (4096/16=256, 8192/512=16)
    dim3 grid(B / 16, K / 512);
    gains_wmma_kernel<<<grid, 32, 0, stream>>>(g, gains, K);

    pair_loss_kernel<<<B, 256, 0, stream>>>(norms, gains, acc, B);

    finalize_kernel<<<1, 1, 0, stream>>>(acc, out);
}